// DLRMModel_21122649161846
// MI455X (gfx1250) — compile-verified
//
#include <hip/hip_runtime.h>
#include <hip/hip_bf16.h>

// ---------------- problem constants (from reference) ----------------
#define NFIELD  26
#define NF      27          // NUM_FEATS
#define DD      64          // embedding dim
#define BATCH   16384
#define VOCAB   100000
#define STILE   16          // samples per workgroup (one WMMA M tile)
#define NI      351         // upper-triangle interactions
#define BM1_    128
#define BM2_    64
#define TOPIN   415
#define TM1_    256

// ---------------- LDS strides (bank-conflict padding, 64 banks) -----
#define FSTR    68          // feats row stride   (68 % 64 = 4  -> conflict-free row walks)
#define CSTR    420         // concat row stride  (420 % 64 = 36, distinct banks over 16 rows)
#define HSTR    132         // h row stride       (132 % 64 = 4)

typedef float v2f __attribute__((ext_vector_type(2)));
typedef float v8f __attribute__((ext_vector_type(8)));

// D = A(16x4) * B(4x16) + C, all fp32 (V_WMMA_F32_16X16X4_F32)
static __device__ __forceinline__ v8f wmma_f32_k4(v2f a, v2f b, v8f c) {
    return __builtin_amdgcn_wmma_f32_16x16x4_f32(false, a, false, b, (short)0, c,
                                                 false, false);
}

// ---------------- kernel 0: fold the rank-1 bottom layer ------------
// dense_embed = price*Wp + bp (rank-1)  =>  h = relu(price*v1 + c1),
// v1 = Wp @ W1, c1 = bp @ W1 + b1. Tiny; one block.
__global__ void dlrm_precompute(const float* __restrict__ Wp,
                                const float* __restrict__ bp,
                                const float* __restrict__ W1,
                                const float* __restrict__ b1,
                                float* __restrict__ v1c1) {
    int j = threadIdx.x;                      // 0..127
    float v = 0.f, c = 0.f;
    for (int d = 0; d < DD; ++d) {
        float w = W1[d * BM1_ + j];
        v = fmaf(Wp[d], w, v);
        c = fmaf(bp[d], w, c);
    }
    v1c1[j]        = v;
    v1c1[BM1_ + j] = c + b1[j];
}

// ---------------- kernel 1: fully fused per-16-sample tile ----------
// LDS ~150 KB -> 2 workgroups (16 wave32) per WGP for gather latency hiding.
__global__ __launch_bounds__(256)
void dlrm_fused(const int*   __restrict__ x_cat,   // [26][16384]
                const float* __restrict__ price,   // [16384]
                const float* __restrict__ emb,     // [26][100000][64]
                const float* __restrict__ Wp,      // [64]
                const float* __restrict__ bp,      // [64]
                const float* __restrict__ W2,      // [128][64]
                const float* __restrict__ b2,      // [64]
                const float* __restrict__ Wt1,     // [415][256]
                const float* __restrict__ bt1,     // [256]
                const float* __restrict__ Wt2,     // [256]
                const float* __restrict__ bt2,     // [1]
                const float* __restrict__ v1c1,    // ws: v1[128], c1[128]
                float* __restrict__ out) {         // [16384]
    __shared__ float lds_feats[STILE * NF * FSTR]; // 114.8 KB
    __shared__ float lds_concat[STILE * CSTR];     // 26.3 KB (415 used + zero pad)
    __shared__ float lds_h[STILE * HSTR];          // 8.3 KB
    __shared__ float lds_zero[DD];                 // zero row for padded gram rows
    __shared__ float lds_outacc[STILE];            // final per-sample accumulators

    const int tid  = threadIdx.x;
    const int lane = tid & 31;
    const int wv   = tid >> 5;                 // wave id, 0..7 (wave32)
    const int b0   = blockIdx.x * STILE;

    // ---- zero concat (incl. K-pad cols 415..419), zero row, accumulators ----
    for (int i = tid; i < STILE * CSTR; i += 256) lds_concat[i] = 0.f;
    if (tid < DD)    lds_zero[tid]   = 0.f;
    if (tid < STILE) lds_outacc[tid] = 0.f;

    // ---- gather: 16 samples x 27 feature rows x 64 floats (float4, coalesced)
    for (int i = tid; i < STILE * NF * (DD / 4); i += 256) {
        int q   = i & 15;                      // float4 within row
        int row = i >> 4;                      // 0..431
        int f   = row % NF;
        int s   = row / NF;
        int b   = b0 + s;
        float4 val;
        if (f < NFIELD) {
            int idx = x_cat[f * BATCH + b];
            const float4* src =
                (const float4*)(emb + ((size_t)f * VOCAB + (size_t)idx) * DD);
            val = src[q];
        } else {                               // dense_embed = price*Wp + bp
            float p = price[b];
            float4 w  = ((const float4*)Wp)[q];
            float4 bb = ((const float4*)bp)[q];
            val.x = fmaf(p, w.x, bb.x); val.y = fmaf(p, w.y, bb.y);
            val.z = fmaf(p, w.z, bb.z); val.w = fmaf(p, w.w, bb.w);
        }
        float* dst = &lds_feats[(s * NF + f) * FSTR + q * 4];
        dst[0] = val.x; dst[1] = val.y; dst[2] = val.z; dst[3] = val.w;
    }

    // ---- h = relu(price*v1 + c1)  (rank-1 bottom layer folded) ----
    for (int i = tid; i < STILE * BM1_; i += 256) {
        int s = i >> 7, j = i & 127;
        lds_h[s * HSTR + j] = fmaxf(fmaf(price[b0 + s], v1c1[j], v1c1[BM1_ + j]), 0.f);
    }
    __syncthreads();

    // Fragment layouts (wave32):
    //   A 16x4 : lanes 0-15 row m, K={k0,k0+1}; lanes 16-31 row m, K={k0+2,k0+3}
    //   B 4x16 : lanes 0-15 col n, K rows {k0,k0+1}; lanes 16-31 K rows {k0+2,k0+3}
    //   C/D    : vgpr r, lane l -> M = r + 8*(l>>4), N = l&15
    const int mhalf = (lane >> 4);             // 0/1
    const int l15   = (lane & 15);
    const int ko    = mhalf * 2;               // per-lane K sub-offset

    // ================= phase 1: gram (48 jobs) + dense_out (4 jobs) =========
    for (int job = wv; job < 52; job += 8) {
        v8f acc = {};
        if (job < 48) {
            // ---- per-sample gram: G = F * F^T, F is 27x64 (padded to 32) ----
            int s = job / 3, t = job % 3;      // t: 0=(M0,N0) 1=(M0,N1) 2=(M1,N1)
            int mbase = (t == 2) ? 16 : 0;
            int nbase = (t == 0) ? 0 : 16;
            const float* F = &lds_feats[s * NF * FSTR];
            int am = mbase + l15;              // A row (feature index)
            int bn = nbase + l15;              // B col = F row
            // OOB (padded) rows read a shared zero row: no per-iter selects,
            // loads merge to ds_load_2addr_b64
            const float* Arow = ((am < NF) ? &F[am * FSTR] : lds_zero) + ko;
            const float* Brow = ((bn < NF) ? &F[bn * FSTR] : lds_zero) + ko;
            #pragma unroll
            for (int k0 = 0; k0 < DD; k0 += 4) {
                v2f a = *(const v2f*)(Arow + k0);
                v2f b = *(const v2f*)(Brow + k0);
                acc = wmma_f32_k4(a, b, acc);
            }
            // scatter strict-upper-triangle entries -> concat[s][triu(i,j)]
            int jj = nbase + l15;
            int m0 = mbase + mhalf * 8;
            #pragma unroll
            for (int r = 0; r < 8; ++r) {
                int ii = m0 + r;
                if (ii < NF && jj < NF && jj > ii) {
                    int flat = 26 * ii - (ii * (ii - 1)) / 2 + (jj - ii - 1);
                    lds_concat[s * CSTR + flat] = acc[r];
                }
            }
        } else {
            // ---- dense_out = relu(h @ W2 + b2): M=16 samples, N tile, K=128 --
            // W2 (32 KB) is L2-resident and shared by all blocks: read global.
            int nn = (job - 48) * 16 + l15;
            const float* Arow = &lds_h[l15 * HSTR] + ko;
            const float* Bcol = W2 + nn;
            #pragma unroll
            for (int k0 = 0; k0 < BM1_; k0 += 4) {
                v2f a = *(const v2f*)(Arow + k0);
                v2f b;
                b.x = Bcol[(k0 + ko) * BM2_];
                b.y = Bcol[(k0 + ko + 1) * BM2_];
                acc = wmma_f32_k4(a, b, acc);
            }
            float bias = b2[nn];
            int m0 = mhalf * 8;
            #pragma unroll
            for (int r = 0; r < 8; ++r)
                lds_concat[(m0 + r) * CSTR + NI + nn] = fmaxf(acc[r] + bias, 0.f);
        }
    }
    __syncthreads();

    // ========== phase 2: t = relu(concat @ Wt1 + bt1);  out += t @ Wt2 ======
    // M=16 samples, N=256 (16 tiles / 8 waves), K=415 -> 412 unguarded + tail.
    for (int nt = wv; nt < 16; nt += 8) {
        int nn = nt * 16 + l15;
        v8f acc = {};
        const float* Arow = &lds_concat[l15 * CSTR] + ko;
        const float* Bcol = Wt1 + nn;
        #pragma unroll 4
        for (int k0 = 0; k0 < 412; k0 += 4) {
            v2f a = *(const v2f*)(Arow + k0);
            v2f b;
            b.x = Bcol[(k0 + ko) * TM1_];
            b.y = Bcol[(k0 + ko + 1) * TM1_];
            acc = wmma_f32_k4(a, b, acc);
        }
        {   // tail K step: rows 412..414 valid, row 415 is zero
            v2f a = *(const v2f*)(Arow + 412);   // concat col 415 is zero-padded
            int ka = 412 + ko;                   // 412 or 414
            v2f b;
            b.x = Bcol[ka * TM1_];
            b.y = (ka + 1 < TOPIN) ? Bcol[(ka + 1) * TM1_] : 0.f;
            acc = wmma_f32_k4(a, b, acc);
        }
        // fold bt1 + relu + Wt2 weight, then reduce the 16 N-lanes per row in
        // registers; lanes 0 and 16 accumulate rows {0..7} / {8..15} into LDS.
        float bias = bt1[nn];
        float wt2n = Wt2[nn];
        float contrib[8];
        #pragma unroll
        for (int r = 0; r < 8; ++r)
            contrib[r] = fmaxf(acc[r] + bias, 0.f) * wt2n;
        #pragma unroll
        for (int off = 1; off < 16; off <<= 1) {
            #pragma unroll
            for (int r = 0; r < 8; ++r)
                contrib[r] += __shfl_xor(contrib[r], off, 32);
        }
        if (l15 == 0) {
            #pragma unroll
            for (int r = 0; r < 8; ++r)
                atomicAdd(&lds_outacc[mhalf * 8 + r], contrib[r]);
        }
    }
    __syncthreads();

    if (tid < STILE) out[b0 + tid] = lds_outacc[tid] + bt2[0];
}

extern "C" void kernel_launch(void* const* d_in, const int* in_sizes, int n_in,
                              void* d_out, int out_size, void* d_ws, size_t ws_size,
                              hipStream_t stream) {
    const int*   x_cat = (const int*)  d_in[0];
    const float* price = (const float*)d_in[1];
    const float* emb   = (const float*)d_in[2];
    const float* Wp    = (const float*)d_in[3];
    const float* bp    = (const float*)d_in[4];
    const float* W1    = (const float*)d_in[5];
    const float* b1    = (const float*)d_in[6];
    const float* W2    = (const float*)d_in[7];
    const float* b2    = (const float*)d_in[8];
    const float* Wt1   = (const float*)d_in[9];
    const float* bt1   = (const float*)d_in[10];
    const float* Wt2   = (const float*)d_in[11];
    const float* bt2   = (const float*)d_in[12];
    float* out  = (float*)d_out;
    float* v1c1 = (float*)d_ws;                 // 256 floats

    dlrm_precompute<<<1, BM1_, 0, stream>>>(Wp, bp, W1, b1, v1c1);
    dlrm_fused<<<BATCH / STILE, 256, 0, stream>>>(x_cat, price, emb, Wp, bp,
                                                  W2, b2, Wt1, bt1, Wt2, bt2,
                                                  v1c1, out);
}